// RTM_62947040690618
// MI455X (gfx1250) — compile-verified
//
#include <hip/hip_runtime.h>
#include <math.h>

// Problem constants (reference: A=32, B=2048, C=256, TK=3, HA=8, DH=32)
#define NBATCH 32
#define SEQ    2048
#define CH     256
#define NTOK   (NBATCH * (SEQ + 1))   // 65568 tokens, divisible by 32
#define NHEAD  8
#define DHEAD  32

#define KC     16                     // K-chunk staged per pipeline stage
#define LDSW   20                     // padded LDS row stride (floats): 80B,
                                      // 16B-aligned for b128, 64-bank-perfect
                                      // for the ds_load_b64 fragment pattern

typedef __attribute__((ext_vector_type(2))) float v2f;
typedef __attribute__((ext_vector_type(8))) float v8f;

// Async global -> LDS copy, 16B per lane, tracked by ASYNCcnt (bypasses VGPRs).
__device__ __forceinline__ void async_b128(uint32_t lds_byte_off, const float* g)
{
  asm volatile("global_load_async_to_lds_b128 %0, %1, off"
               :: "v"(lds_byte_off), "v"(g) : "memory");
}

// ---------------------------------------------------------------------------
// WMMA GEMM:  D[M x N](ldd) = A[M x K] * W[N x K]^T  (+bias[N]) (+resid) (+gelu)
//
// Block = 128 threads (4 waves) -> 32 rows x 256 cols of D.
// Wave w covers cols w*64..w*64+63 with 2 row-tiles x 4 col-tiles of
// v_wmma_f32_16x16x4_f32 accumulators (full fp32 precision on the matrix pipe).
// K is consumed in KC=16 chunks, double-buffered in LDS and filled with
// global_load_async_to_lds_b128; the inner loop reads bank-conflict-free
// ds_load_b64 fragments, so global latency never serializes the WMMAs.
// Residual / bias fold into accumulator init; GELU applied on accumulators.
// ---------------------------------------------------------------------------
template <int HAS_BIAS, int HAS_RESID, int DO_GELU>
__global__ __launch_bounds__(128) void wmma_gemm_f32(
    const float* __restrict__ Amat,   // M x K
    const float* __restrict__ Wmat,   // N x K (row major)
    const float* __restrict__ biasv,  // N       (or nullptr)
    const float* __restrict__ resid,  // M x ldd (or nullptr)
    float* __restrict__ Dmat,         // M x ldd
    int K, int ldd)
{
  __shared__ float Wb[2][256 * LDSW];   // 40960 B
  __shared__ float Ab[2][32 * LDSW];    //  5120 B

  const int tid   = threadIdx.x;
  const int lane  = tid & 31;
  const int wave  = tid >> 5;
  const int ln    = lane & 15;   // col within 16-tile / A-row within tile
  const int khalf = lane >> 4;   // which K-pair this lane holds

  const int m0  = blockIdx.x * 32;   // block row base
  const int nb0 = blockIdx.y * 256;  // block col base
  const int nw  = wave * 64;         // wave col offset within block

  // ---- accumulator init (residual / bias fused) ----
  v8f acc[2][4];
#pragma unroll
  for (int rt = 0; rt < 2; ++rt)
#pragma unroll
    for (int t = 0; t < 4; ++t) {
      float cb = 0.0f;
      if (HAS_BIAS) cb = biasv[nb0 + nw + t * 16 + ln];
#pragma unroll
      for (int r = 0; r < 8; ++r) {
        float c = cb;
        if (HAS_RESID)
          c += resid[(size_t)(m0 + rt * 16 + khalf * 8 + r) * ldd +
                     nb0 + nw + t * 16 + ln];
        acc[rt][t][r] = c;
      }
    }

  // ---- async staging: one K-chunk = W 256x16 (8 xfers/thd) + A 32x16 (1/thd)
  auto stage = [&](int buf, int kc) {
#pragma unroll
    for (int it = 0; it < 8; ++it) {
      int q = it * 128 + tid;          // 0..1023 : W row q>>2, segment q&3
      int row = q >> 2, seg = q & 3;
      async_b128((uint32_t)(uintptr_t)&Wb[buf][row * LDSW + seg * 4],
                 Wmat + (size_t)(nb0 + row) * K + kc + seg * 4);
    }
    {
      int row = tid >> 2, seg = tid & 3;   // 0..127 : A rows 0..31
      async_b128((uint32_t)(uintptr_t)&Ab[buf][row * LDSW + seg * 4],
                 Amat + (size_t)(m0 + row) * K + kc + seg * 4);
    }
  };

  stage(0, 0);
  const int nchunks = K / KC;
  for (int c = 0; c < nchunks; ++c) {
    const int cur = c & 1;
    if (c + 1 < nchunks) {
      stage(cur ^ 1, (c + 1) * KC);
      // 9 asyncs just issued may stay in flight; oldest 9 (chunk c) complete
      asm volatile("s_wait_asynccnt 0x9" ::: "memory");
    } else {
      asm volatile("s_wait_asynccnt 0x0" ::: "memory");
    }
    __syncthreads();

#pragma unroll
    for (int kk = 0; kk < KC; kk += 4) {
      v2f a0 = *(const v2f*)&Ab[cur][(ln)      * LDSW + kk + khalf * 2];
      v2f a1 = *(const v2f*)&Ab[cur][(16 + ln) * LDSW + kk + khalf * 2];
#pragma unroll
      for (int t = 0; t < 4; ++t) {
        v2f b = *(const v2f*)&Wb[cur][(nw + t * 16 + ln) * LDSW + kk + khalf * 2];
        acc[0][t] = __builtin_amdgcn_wmma_f32_16x16x4_f32(
            false, a0, false, b, (short)0, acc[0][t], false, false);
        acc[1][t] = __builtin_amdgcn_wmma_f32_16x16x4_f32(
            false, a1, false, b, (short)0, acc[1][t], false, false);
      }
    }
    __syncthreads();
  }

  // ---- epilogue ----
#pragma unroll
  for (int rt = 0; rt < 2; ++rt)
#pragma unroll
    for (int t = 0; t < 4; ++t)
#pragma unroll
      for (int r = 0; r < 8; ++r) {
        float v = acc[rt][t][r];
        if (DO_GELU)  // exact gelu: 0.5*x*(1+erf(x/sqrt(2)))
          v = 0.5f * v * (1.0f + erff(v * 0.70710678118654752f));
        Dmat[(size_t)(m0 + rt * 16 + khalf * 8 + r) * ldd +
             nb0 + nw + t * 16 + ln] = v;
      }
}

// ---------------------------------------------------------------------------
// Assemble savespace = concat(cls, tmp) + bias  (bias flat layout == S layout)
// ---------------------------------------------------------------------------
__global__ void assemble_k(const float* __restrict__ tmp,   // (32*2048) x 256
                           const float* __restrict__ cls,   // 32 x 256
                           const float* __restrict__ bias,  // NTOK x 256
                           float* __restrict__ S)           // NTOK x 256
{
  size_t idx = (size_t)blockIdx.x * blockDim.x + threadIdx.x;
  if (idx >= (size_t)NTOK * CH) return;
  int    c   = (int)(idx & (CH - 1));
  size_t tok = idx >> 8;
  int    i   = (int)(tok / (SEQ + 1));
  int    j   = (int)(tok % (SEQ + 1));
  float  v   = (j == 0) ? cls[i * CH + c]
                        : tmp[((size_t)i * SEQ + (j - 1)) * CH + c];
  S[idx] = v + bias[idx];
}

// ---------------------------------------------------------------------------
// LayerNorm over C=256: one wave per token, 8 floats per lane, shfl butterfly.
// ---------------------------------------------------------------------------
__global__ __launch_bounds__(256) void layernorm_k(
    const float* __restrict__ x, const float* __restrict__ g,
    const float* __restrict__ b, float* __restrict__ out)
{
  int gw   = (int)((blockIdx.x * (size_t)blockDim.x + threadIdx.x) >> 5);
  int lane = threadIdx.x & 31;
  if (gw >= NTOK) return;
  const float* row = x + (size_t)gw * CH + lane * 8;
  float4 u0 = *(const float4*)(row);
  float4 u1 = *(const float4*)(row + 4);
  float s  = u0.x + u0.y + u0.z + u0.w + u1.x + u1.y + u1.z + u1.w;
  float sq = u0.x*u0.x + u0.y*u0.y + u0.z*u0.z + u0.w*u0.w +
             u1.x*u1.x + u1.y*u1.y + u1.z*u1.z + u1.w*u1.w;
#pragma unroll
  for (int off = 16; off; off >>= 1) {
    s  += __shfl_xor(s, off, 32);
    sq += __shfl_xor(sq, off, 32);
  }
  float mu  = s * (1.0f / CH);
  float var = sq * (1.0f / CH) - mu * mu;
  float inv = rsqrtf(var + 1e-5f);
  float4 g0 = *(const float4*)(g + lane * 8);
  float4 g1 = *(const float4*)(g + lane * 8 + 4);
  float4 b0 = *(const float4*)(b + lane * 8);
  float4 b1 = *(const float4*)(b + lane * 8 + 4);
  float4 o0, o1;
  o0.x = (u0.x - mu) * inv * g0.x + b0.x;  o0.y = (u0.y - mu) * inv * g0.y + b0.y;
  o0.z = (u0.z - mu) * inv * g0.z + b0.z;  o0.w = (u0.w - mu) * inv * g0.w + b0.w;
  o1.x = (u1.x - mu) * inv * g1.x + b1.x;  o1.y = (u1.y - mu) * inv * g1.y + b1.y;
  o1.z = (u1.z - mu) * inv * g1.z + b1.z;  o1.w = (u1.w - mu) * inv * g1.w + b1.w;
  float* orow = out + (size_t)gw * CH + lane * 8;
  *(float4*)(orow)     = o0;
  *(float4*)(orow + 4) = o1;
}

// ---------------------------------------------------------------------------
// Attention scalar: one wave per (token, head); DH=32 == one lane per d.
// imv[tok, h*32+d] = (sum_d q*k) / sqrt(DH) * v
// ---------------------------------------------------------------------------
__global__ __launch_bounds__(256) void attn_k(
    const float* __restrict__ qkv,  // NTOK x 768  (q | k | v)
    float* __restrict__ imv)        // NTOK x 256
{
  int gw   = (int)((blockIdx.x * (size_t)blockDim.x + threadIdx.x) >> 5);
  int lane = threadIdx.x & 31;
  int tok  = gw >> 3;
  int h    = gw & 7;
  if (tok >= NTOK) return;
  const float* base = qkv + (size_t)tok * (3 * CH) + h * DHEAD + lane;
  float q  = base[0];
  float kk = base[CH];
  float v  = base[2 * CH];
  float d  = q * kk;
#pragma unroll
  for (int off = 16; off; off >>= 1) d += __shfl_xor(d, off, 32);
  imv[(size_t)tok * CH + h * DHEAD + lane] = d * 0.17677669529663687f * v; // 1/sqrt(32)
}

// ---------------------------------------------------------------------------
// In-place inclusive cumsum over j = 0..SEQ-1 (position SEQ untouched).
// 32 blocks x 256 threads: thread = one channel of one batch; fully coalesced
// 1KB line per step per block.
// ---------------------------------------------------------------------------
__global__ __launch_bounds__(256) void scan_k(float* __restrict__ imv)
{
  float* p = imv + ((size_t)blockIdx.x * (SEQ + 1)) * CH + threadIdx.x;
  float acc = 0.0f;
  for (int j = 0; j < SEQ; ++j) {
    acc += *p;
    *p = acc;
    p += CH;
  }
}

// ---------------------------------------------------------------------------
extern "C" void kernel_launch(void* const* d_in, const int* in_sizes, int n_in,
                              void* d_out, int out_size, void* d_ws, size_t ws_size,
                              hipStream_t stream)
{
  const float* x      = (const float*)d_in[0];   // (32, 2048, 256)
  const float* weight = (const float*)d_in[1];   // (256, 256)
  const float* bias   = (const float*)d_in[2];   // (32, 2049, 256)
  const float* cls    = (const float*)d_in[3];   // (32, 1, 256)
  const float* Wqkv   = (const float*)d_in[4];   // (3, 3, 8, 32, 256)
  const float* Wo     = (const float*)d_in[5];   // (3, 256, 256)
  const float* ln1g   = (const float*)d_in[6];
  const float* ln1b   = (const float*)d_in[7];
  const float* ln2g   = (const float*)d_in[8];
  const float* ln2b   = (const float*)d_in[9];
  const float* fc1w   = (const float*)d_in[10];  // (1024, 256)
  const float* fc1b   = (const float*)d_in[11];  // (1024,)
  const float* fc2w   = (const float*)d_in[12];  // (256, 1024)
  const float* fc2b   = (const float*)d_in[13];  // (256,)

  float* S  = (float*)d_out;                 // savespace lives in d_out: NTOK x 256
  float* R1 = (float*)d_ws;                  // NTOK x 1024: embed-tmp -> qkv -> h
  float* R2 = R1 + (size_t)NTOK * 1024;      // NTOK x 256 : z -> imv -> z2

  const dim3 blk(128);

  // 1) embed GEMM: tmp = x @ weight^T   (M=65536, N=256, K=256)
  wmma_gemm_f32<0, 0, 0><<<dim3((NBATCH * SEQ) / 32, 1), blk, 0, stream>>>(
      x, weight, nullptr, nullptr, R1, CH, CH);

  // 2) savespace = concat(cls, tmp) + bias
  {
    size_t total = (size_t)NTOK * CH;
    assemble_k<<<(unsigned)((total + 255) / 256), 256, 0, stream>>>(R1, cls, bias, S);
  }

  for (int a = 0; a < 3; ++a) {
    const float* wq = Wqkv + (size_t)a * 3 * NHEAD * DHEAD * CH;  // (768, 256)
    const float* wo = Wo + (size_t)a * CH * CH;                   // (256, 256)

    // z = LN1(S)
    layernorm_k<<<NTOK / 8, 256, 0, stream>>>(S, ln1g, ln1b, R2);
    // qkv = z @ Wqkv^T   (N=768)
    wmma_gemm_f32<0, 0, 0><<<dim3(NTOK / 32, 3), blk, 0, stream>>>(
        R2, wq, nullptr, nullptr, R1, CH, 3 * CH);
    // imv = (q.k)/sqrt(DH) * v
    attn_k<<<NTOK, 256, 0, stream>>>(R1, R2);
    // cumsum over sequence (last position untouched)
    scan_k<<<NBATCH, 256, 0, stream>>>(R2);
    // S = imv @ Wo^T + S
    wmma_gemm_f32<0, 1, 0><<<dim3(NTOK / 32, 1), blk, 0, stream>>>(
        R2, wo, nullptr, S, S, CH, CH);
    // z2 = LN2(S)
    layernorm_k<<<NTOK / 8, 256, 0, stream>>>(S, ln2g, ln2b, R2);
    // h = gelu(z2 @ fc1^T + b1)   (N=1024)
    wmma_gemm_f32<1, 0, 1><<<dim3(NTOK / 32, 4), blk, 0, stream>>>(
        R2, fc1w, fc1b, nullptr, R1, CH, 4 * CH);
    // S = h @ fc2^T + b2 + S   (K=1024)
    wmma_gemm_f32<1, 1, 0><<<dim3(NTOK / 32, 1), blk, 0, stream>>>(
        R1, fc2w, fc2b, S, S, 4 * CH, CH);
  }
}